// LRU_27711128994526
// MI455X (gfx1250) — compile-verified
//
#include <hip/hip_runtime.h>
#include <math.h>

typedef __attribute__((ext_vector_type(2))) float v2f;
typedef __attribute__((ext_vector_type(8))) float v8f;

#define T_LEN   2048
#define B_LEN   16
#define D_LEN   256
#define M_LEN   (T_LEN * B_LEN)          // 32768 rows (t*B + b)
#define CHUNK   128
#define NCHUNK  (T_LEN / CHUNK)          // 16
#define CH      (B_LEN * D_LEN)          // 4096 scan channels
#define EPS_RE  1e-10f

// ---------------------------------------------------------------------------
// 1) Deinterleave complex weights into planar re/im. Store -Im(C_w) because
//    f32 WMMA NEG bits only support C-negate (ISA 7.12), so we bake the minus
//    sign for Re(h*C) = h_re*Cre - h_im*Cim into the weight plane.
// ---------------------------------------------------------------------------
__global__ void lru_prep_weights(const float2* __restrict__ Bw,
                                 const float2* __restrict__ Cw,
                                 float* __restrict__ Bre, float* __restrict__ Bim,
                                 float* __restrict__ Cre, float* __restrict__ nCim) {
    int i = blockIdx.x * blockDim.x + threadIdx.x;   // < D*D = 65536
    float2 b = Bw[i];
    Bre[i] = b.x;
    Bim[i] = b.y;
    float2 c = Cw[i];
    Cre[i] = c.x;
    nCim[i] = -c.y;
}

// ---------------------------------------------------------------------------
// WMMA fragment addressing for V_WMMA_F32_16X16X4_F32 (ISA 7.12.2):
//   A (16x4):  lane L in [0,15]: row M=L, K={k0,k0+1}; lanes [16,31]: K={k0+2,k0+3}
//   B (4x16):  symmetric: lane low nibble = N column, lane[4] selects K pair
//   C/D:       VGPR r, lanes 0-15 -> M=r, N=lane; lanes 16-31 -> M=8+r, N=lane-16
// All per-lane offsets are 32-bit and loop-invariant; the K index goes into the
// instruction's immediate offset (full unroll), so the hot loop is pure
// global_load_b64 (SADDR form) + v_wmma with no address VALU.
// ---------------------------------------------------------------------------

// 2) bx = (x @ B_w^T) + EPS  (complex out, planar). Wave computes 32x32 tile,
//    re+im accumulators (8 WMMAs per K step of 4, K fully unrolled).
__global__ __launch_bounds__(128) void lru_gemm_bx(
    const float* __restrict__ x,
    const float* __restrict__ Bre, const float* __restrict__ Bim,
    float* __restrict__ out_re, float* __restrict__ out_im) {
    const int lane = threadIdx.x & 31;
    const int wave = threadIdx.x >> 5;
    const int tile = blockIdx.x * 4 + wave;
    const int NT   = D_LEN / 32;                  // 8 N-tiles
    const int M0   = (tile / NT) * 32;
    const int N0   = (tile % NT) * 32;
    const int lq   = lane & 15;
    const int half = lane >> 4;
    const int koff = half * 2;

    v8f z = {0.f, 0.f, 0.f, 0.f, 0.f, 0.f, 0.f, 0.f};
    v8f ar00 = z, ar01 = z, ar10 = z, ar11 = z;
    v8f ai00 = z, ai01 = z, ai10 = z, ai11 = z;

    // 32-bit loop-invariant per-lane element offsets (base ptrs stay uniform)
    const int a0o = (M0 + lq)      * D_LEN + koff;
    const int a1o = (M0 + 16 + lq) * D_LEN + koff;
    const int w0o = (N0 + lq)      * D_LEN + koff;
    const int w1o = (N0 + 16 + lq) * D_LEN + koff;

    #pragma unroll
    for (int k0 = 0; k0 < D_LEN; k0 += 4) {
        v2f a0  = *(const v2f*)(x   + a0o + k0);
        v2f a1  = *(const v2f*)(x   + a1o + k0);
        v2f br0 = *(const v2f*)(Bre + w0o + k0);
        v2f br1 = *(const v2f*)(Bre + w1o + k0);
        v2f bi0 = *(const v2f*)(Bim + w0o + k0);
        v2f bi1 = *(const v2f*)(Bim + w1o + k0);
        ar00 = __builtin_amdgcn_wmma_f32_16x16x4_f32(false, a0, false, br0, (short)0, ar00, false, false);
        ar01 = __builtin_amdgcn_wmma_f32_16x16x4_f32(false, a0, false, br1, (short)0, ar01, false, false);
        ar10 = __builtin_amdgcn_wmma_f32_16x16x4_f32(false, a1, false, br0, (short)0, ar10, false, false);
        ar11 = __builtin_amdgcn_wmma_f32_16x16x4_f32(false, a1, false, br1, (short)0, ar11, false, false);
        ai00 = __builtin_amdgcn_wmma_f32_16x16x4_f32(false, a0, false, bi0, (short)0, ai00, false, false);
        ai01 = __builtin_amdgcn_wmma_f32_16x16x4_f32(false, a0, false, bi1, (short)0, ai01, false, false);
        ai10 = __builtin_amdgcn_wmma_f32_16x16x4_f32(false, a1, false, bi0, (short)0, ai10, false, false);
        ai11 = __builtin_amdgcn_wmma_f32_16x16x4_f32(false, a1, false, bi1, (short)0, ai11, false, false);
    }

    #pragma unroll
    for (int r = 0; r < 8; ++r) {
        int mA = M0 + r + 8 * half;
        int mB = mA + 16;
        int nA = N0 + lq;
        int nB = nA + 16;
        out_re[mA * D_LEN + nA] = ar00[r] + EPS_RE;
        out_re[mA * D_LEN + nB] = ar01[r] + EPS_RE;
        out_re[mB * D_LEN + nA] = ar10[r] + EPS_RE;
        out_re[mB * D_LEN + nB] = ar11[r] + EPS_RE;
        out_im[mA * D_LEN + nA] = ai00[r];
        out_im[mA * D_LEN + nB] = ai01[r];
        out_im[mB * D_LEN + nA] = ai10[r];
        out_im[mB * D_LEN + nB] = ai11[r];
    }
}

// Helper: per-channel lambda = exp(-exp(nu_log) + i*exp(theta_log))
__device__ __forceinline__ void lru_lambda(const float* nu_log, const float* th_log,
                                           int d, float& lr, float& li) {
    float nu = __expf(nu_log[d]);
    float th = __expf(th_log[d]);
    float mag = __expf(-nu);
    lr = mag * __cosf(th);
    li = mag * __sinf(th);
}

// 3) In-place chunked scan: h = lambda*h + gamma*bx over CHUNK steps (h0 = 0),
//    overwrite buffer with local hidden states, emit per-chunk carry.
__global__ __launch_bounds__(256) void lru_scan_chunk(
    float* __restrict__ buf_re, float* __restrict__ buf_im,
    float* __restrict__ carry_re, float* __restrict__ carry_im,
    const float* __restrict__ nu_log, const float* __restrict__ th_log,
    const float* __restrict__ gamma_log) {
    int tid = blockIdx.x * blockDim.x + threadIdx.x;   // < NCHUNK*CH
    int channel = tid & (CH - 1);
    int c = tid >> 12;                                 // CH == 4096 == 2^12
    int d = channel & (D_LEN - 1);

    float lr, li;
    lru_lambda(nu_log, th_log, d, lr, li);
    float g = __expf(gamma_log[d]);

    float hr = 0.f, hi = 0.f;
    int base = c * CHUNK * CH + channel;
    #pragma unroll 4
    for (int i = 0; i < CHUNK; ++i) {
        float vr = buf_re[base];
        float vi = buf_im[base];
        float nhr = fmaf(lr, hr, fmaf(-li, hi, g * vr));
        float nhi = fmaf(lr, hi, fmaf( li, hr, g * vi));
        hr = nhr; hi = nhi;
        buf_re[base] = hr;
        buf_im[base] = hi;
        base += CH;
    }
    carry_re[c * CH + channel] = hr;
    carry_im[c * CH + channel] = hi;
}

// 4) Combine carries across chunks: H_incl[c] = lam^CHUNK * H_incl[c-1] + carry[c].
//    Emit exclusive prefixes for fix-up and the final hidden state h_{T-1}
//    (second reference output) as interleaved complex64 at d_out + T*B*D.
__global__ __launch_bounds__(256) void lru_carry_combine(
    const float* __restrict__ carry_re, const float* __restrict__ carry_im,
    float* __restrict__ Hx_re, float* __restrict__ Hx_im,
    const float* __restrict__ nu_log, const float* __restrict__ th_log,
    float* __restrict__ out_tail) {
    int channel = blockIdx.x * blockDim.x + threadIdx.x;   // < CH
    int d = channel & (D_LEN - 1);

    float nu = __expf(nu_log[d]);
    float th = __expf(th_log[d]);
    // lambda^CHUNK: magnitude exp(-nu*CHUNK), phase th*CHUNK (reduced in double)
    double ang = fmod((double)th * (double)CHUNK, 6.283185307179586);
    float mag = __expf(-nu * (float)CHUNK);
    float Lr = mag * __cosf((float)ang);
    float Li = mag * __sinf((float)ang);

    float hr = 0.f, hi = 0.f;
    for (int c = 0; c < NCHUNK; ++c) {
        int idx = c * CH + channel;
        Hx_re[idx] = hr;                  // exclusive prefix
        Hx_im[idx] = hi;
        float cr = carry_re[idx];
        float ci = carry_im[idx];
        float nhr = fmaf(Lr, hr, fmaf(-Li, hi, cr));
        float nhi = fmaf(Lr, hi, fmaf( Li, hr, ci));
        hr = nhr; hi = nhi;
    }
    out_tail[2 * channel]     = hr;       // ht[-1] complex64 interleaved
    out_tail[2 * channel + 1] = hi;
}

// 5) Fix-up: h_full[t] = h_local[t] + lambda^{i+1} * H_excl[chunk], i = t in chunk.
__global__ __launch_bounds__(256) void lru_fixup(
    float* __restrict__ buf_re, float* __restrict__ buf_im,
    const float* __restrict__ Hx_re, const float* __restrict__ Hx_im,
    const float* __restrict__ nu_log, const float* __restrict__ th_log) {
    int tid = blockIdx.x * blockDim.x + threadIdx.x;
    int channel = tid & (CH - 1);
    int c = tid >> 12;
    if (c == 0) return;                   // chunk 0 has no prefix
    int d = channel & (D_LEN - 1);

    float lr, li;
    lru_lambda(nu_log, th_log, d, lr, li);
    float Her = Hx_re[c * CH + channel];
    float Hei = Hx_im[c * CH + channel];
    // w = lambda^{i+1} * H_excl, starting at i=0
    float wr = fmaf(lr, Her, -li * Hei);
    float wi = fmaf(lr, Hei,  li * Her);

    int base = c * CHUNK * CH + channel;
    #pragma unroll 4
    for (int i = 0; i < CHUNK; ++i) {
        buf_re[base] += wr;
        buf_im[base] += wi;
        float nwr = fmaf(lr, wr, -li * wi);
        float nwi = fmaf(lr, wi,  li * wr);
        wr = nwr; wi = nwi;
        base += CH;
    }
}

// 6) y = Re(h @ C_w^T) + d_vec * x = h_re@Cre^T + h_im@(-Cim)^T + d_vec*x.
__global__ __launch_bounds__(128) void lru_gemm_y(
    const float* __restrict__ hre, const float* __restrict__ him,
    const float* __restrict__ Cre, const float* __restrict__ nCim,
    const float* __restrict__ x, const float* __restrict__ dvec,
    float* __restrict__ y) {
    const int lane = threadIdx.x & 31;
    const int wave = threadIdx.x >> 5;
    const int tile = blockIdx.x * 4 + wave;
    const int NT   = D_LEN / 32;
    const int M0   = (tile / NT) * 32;
    const int N0   = (tile % NT) * 32;
    const int lq   = lane & 15;
    const int half = lane >> 4;
    const int koff = half * 2;

    v8f z = {0.f, 0.f, 0.f, 0.f, 0.f, 0.f, 0.f, 0.f};
    v8f acc00 = z, acc01 = z, acc10 = z, acc11 = z;

    const int a0o = (M0 + lq)      * D_LEN + koff;
    const int a1o = (M0 + 16 + lq) * D_LEN + koff;
    const int w0o = (N0 + lq)      * D_LEN + koff;
    const int w1o = (N0 + 16 + lq) * D_LEN + koff;

    #pragma unroll
    for (int k0 = 0; k0 < D_LEN; k0 += 4) {
        v2f ar0 = *(const v2f*)(hre  + a0o + k0);
        v2f ar1 = *(const v2f*)(hre  + a1o + k0);
        v2f ai0 = *(const v2f*)(him  + a0o + k0);
        v2f ai1 = *(const v2f*)(him  + a1o + k0);
        v2f cr0 = *(const v2f*)(Cre  + w0o + k0);
        v2f cr1 = *(const v2f*)(Cre  + w1o + k0);
        v2f cn0 = *(const v2f*)(nCim + w0o + k0);
        v2f cn1 = *(const v2f*)(nCim + w1o + k0);
        acc00 = __builtin_amdgcn_wmma_f32_16x16x4_f32(false, ar0, false, cr0, (short)0, acc00, false, false);
        acc00 = __builtin_amdgcn_wmma_f32_16x16x4_f32(false, ai0, false, cn0, (short)0, acc00, false, false);
        acc01 = __builtin_amdgcn_wmma_f32_16x16x4_f32(false, ar0, false, cr1, (short)0, acc01, false, false);
        acc01 = __builtin_amdgcn_wmma_f32_16x16x4_f32(false, ai0, false, cn1, (short)0, acc01, false, false);
        acc10 = __builtin_amdgcn_wmma_f32_16x16x4_f32(false, ar1, false, cr0, (short)0, acc10, false, false);
        acc10 = __builtin_amdgcn_wmma_f32_16x16x4_f32(false, ai1, false, cn0, (short)0, acc10, false, false);
        acc11 = __builtin_amdgcn_wmma_f32_16x16x4_f32(false, ar1, false, cr1, (short)0, acc11, false, false);
        acc11 = __builtin_amdgcn_wmma_f32_16x16x4_f32(false, ai1, false, cn1, (short)0, acc11, false, false);
    }

    float dvA = dvec[N0 + lq];
    float dvB = dvec[N0 + 16 + lq];
    #pragma unroll
    for (int r = 0; r < 8; ++r) {
        int mA = M0 + r + 8 * half;
        int mB = mA + 16;
        int nA = N0 + lq;
        int nB = nA + 16;
        y[mA * D_LEN + nA] = acc00[r] + dvA * x[mA * D_LEN + nA];
        y[mA * D_LEN + nB] = acc01[r] + dvB * x[mA * D_LEN + nB];
        y[mB * D_LEN + nA] = acc10[r] + dvA * x[mB * D_LEN + nA];
        y[mB * D_LEN + nB] = acc11[r] + dvB * x[mB * D_LEN + nB];
    }
}

extern "C" void kernel_launch(void* const* d_in, const int* in_sizes, int n_in,
                              void* d_out, int out_size, void* d_ws, size_t ws_size,
                              hipStream_t stream) {
    const float*  x         = (const float*)d_in[0];
    const float*  nu_log    = (const float*)d_in[1];
    const float*  theta_log = (const float*)d_in[2];
    const float2* B_w       = (const float2*)d_in[3];   // complex64 interleaved
    const float2* C_w       = (const float2*)d_in[4];   // complex64 interleaved
    const float*  d_vec     = (const float*)d_in[5];
    const float*  gamma_log = (const float*)d_in[6];
    float* out = (float*)d_out;
    float* ws  = (float*)d_ws;

    const size_t S  = (size_t)M_LEN * D_LEN;    // 8388608 elements
    const size_t DD = (size_t)D_LEN * D_LEN;    // 65536
    float* buf_re   = ws;                       // bx -> ht, in place
    float* buf_im   = buf_re + S;
    float* Bre      = buf_im + S;
    float* Bim      = Bre + DD;
    float* Cre      = Bim + DD;
    float* nCim     = Cre + DD;
    float* carry_re = nCim + DD;
    float* carry_im = carry_re + (size_t)NCHUNK * CH;
    float* Hx_re    = carry_im + (size_t)NCHUNK * CH;
    float* Hx_im    = Hx_re + (size_t)NCHUNK * CH;

    const int gemm_blocks = (M_LEN / 32) * (D_LEN / 32) / 4;   // 2048 (4 waves/block)

    lru_prep_weights<<<DD / 256, 256, 0, stream>>>(B_w, C_w, Bre, Bim, Cre, nCim);
    lru_gemm_bx<<<gemm_blocks, 128, 0, stream>>>(x, Bre, Bim, buf_re, buf_im);
    lru_scan_chunk<<<NCHUNK * CH / 256, 256, 0, stream>>>(buf_re, buf_im, carry_re, carry_im,
                                                          nu_log, theta_log, gamma_log);
    lru_carry_combine<<<CH / 256, 256, 0, stream>>>(carry_re, carry_im, Hx_re, Hx_im,
                                                    nu_log, theta_log, out + S);
    lru_fixup<<<NCHUNK * CH / 256, 256, 0, stream>>>(buf_re, buf_im, Hx_re, Hx_im,
                                                     nu_log, theta_log);
    lru_gemm_y<<<gemm_blocks, 128, 0, stream>>>(buf_re, buf_im, Cre, nCim, x, d_vec, out);
}